// NeighborSampler_14559939133495
// MI455X (gfx1250) — compile-verified
//
#include <hip/hip_runtime.h>
#include <stdint.h>

#define D 128                 // MAX_DEGREE, fixed by the reference
#define ROWS_PER_BLOCK 2
#define THREADS (D * ROWS_PER_BLOCK)

// ---- CDNA5 async global->LDS copy path -------------------------------------
#if defined(__gfx1250__)
typedef __attribute__((address_space(1))) int gbl_int_t;   // prints as __device__ int*
typedef __attribute__((address_space(3))) int lds_int_t;   // prints as __shared__ int*

#if __has_builtin(__builtin_amdgcn_global_load_async_to_lds_b32)
#define ASYNC_COPY_B32(gptr, lptr)                                            \
    __builtin_amdgcn_global_load_async_to_lds_b32(                            \
        (gbl_int_t*)(gptr), (lds_int_t*)(lptr), 0, 0)
#else
// VGLOBAL "GV" form: dst = LDS byte address (wave-relative), src = 64-bit VA.
#define ASYNC_COPY_B32(gptr, lptr)                                            \
    asm volatile("global_load_async_to_lds_b32 %0, %1, off"                   \
                 :: "v"((unsigned)(uintptr_t)(lptr)),                         \
                    "v"((unsigned long long)(uintptr_t)(gptr))                \
                 : "memory")
#endif

#if __has_builtin(__builtin_amdgcn_s_wait_asynccnt)
#define WAIT_ASYNC() __builtin_amdgcn_s_wait_asynccnt(0)
#else
#define WAIT_ASYNC() asm volatile("s_wait_asynccnt 0x0" ::: "memory")
#endif

#define USE_ASYNC_LDS 1
#else
#define USE_ASYNC_LDS 0
#endif
// ----------------------------------------------------------------------------

__global__ __launch_bounds__(THREADS)
void NeighborSampler_14559939133495_kernel(const int* __restrict__ ids,
                                           const int* __restrict__ num_samples_p,
                                           const int* __restrict__ adj_info,
                                           const float* __restrict__ centrality,
                                           int* __restrict__ out,
                                           int batch, int num_nodes)
{
    __shared__ int                lds_adj[ROWS_PER_BLOCK][D];
    __shared__ unsigned long long keys[ROWS_PER_BLOCK][D];

    const int tid = (int)(threadIdx.x & (D - 1));   // position within row
    const int sub = (int)(threadIdx.x >> 7);        // row slot within block
    const int row = (int)blockIdx.x * ROWS_PER_BLOCK + sub;
    const bool valid = (row < batch);
    const int  crow  = valid ? row : (batch - 1);   // keep all lanes uniform

    const int id = ids[crow];
    const long long gidx = (long long)id * D + tid;

#if USE_ASYNC_LDS
    // One wave32 instruction moves 128B of the adjacency row into LDS.
    ASYNC_COPY_B32(adj_info + gidx, &lds_adj[sub][tid]);
    WAIT_ASYNC();
#else
    lds_adj[sub][tid] = adj_info[gidx];
#endif
    __syncthreads();

    // clip to [0, num_nodes-1] (matches jnp.clip in the reference)
    int nb = lds_adj[sub][tid];
    nb = nb < 0 ? 0 : (nb > num_nodes - 1 ? num_nodes - 1 : nb);
    lds_adj[sub][tid] = nb;   // visible before first sort barrier

    // centrality in [0,1): IEEE bits are order-preserving for non-negative
    // floats. Key = score(desc) with lower-original-index-first tie-break.
    const unsigned sbits = __float_as_uint(centrality[nb]);
    keys[sub][tid] = ((unsigned long long)sbits << 32) |
                     (unsigned)(D - 1 - tid);

    // Bitonic sort, descending, 128 elements / 128 threads, 28 steps.
    for (int k = 2; k <= D; k <<= 1) {
        for (int j = k >> 1; j > 0; j >>= 1) {
            __syncthreads();
            if ((tid & j) == 0) {
                const int partner = tid | j;
                const unsigned long long a = keys[sub][tid];
                const unsigned long long b = keys[sub][partner];
                const bool desc_blk = ((tid & k) == 0);
                const bool sw = desc_blk ? (a < b) : (a > b);
                if (sw) { keys[sub][tid] = b; keys[sub][partner] = a; }
            }
        }
    }
    __syncthreads();

    const int ns = *num_samples_p;
    if (valid && tid < ns) {
        const int orig = D - 1 - (int)(keys[sub][tid] & (D - 1));
        out[(long long)row * ns + tid] = lds_adj[sub][orig];
    }
}

extern "C" void kernel_launch(void* const* d_in, const int* in_sizes, int n_in,
                              void* d_out, int out_size, void* d_ws, size_t ws_size,
                              hipStream_t stream)
{
    (void)n_in; (void)out_size; (void)d_ws; (void)ws_size;
    const int*   ids  = (const int*)d_in[0];
    const int*   nsp  = (const int*)d_in[1];   // scalar num_samples on device
    const int*   adj  = (const int*)d_in[2];
    const float* cent = (const float*)d_in[3];
    int*         out  = (int*)d_out;           // reference output dtype: int32

    const int batch     = in_sizes[0];
    const int num_nodes = in_sizes[3];

    const int blocks = (batch + ROWS_PER_BLOCK - 1) / ROWS_PER_BLOCK;
    NeighborSampler_14559939133495_kernel<<<blocks, THREADS, 0, stream>>>(
        ids, nsp, adj, cent, out, batch, num_nodes);
}